// SLSTMHead_3006477107697
// MI455X (gfx1250) — compile-verified
//
#include <hip/hip_runtime.h>

// SLSTM head: B=1024 rows, H=1024 hidden, S=4096 steps, out (B,S,2) f32.
// One block per batch row, 1024 threads (32 waves), thread = hidden index h.
// Recurrence fully in registers (log2-domain normalizer, prescaled weights);
// projection via V_WMMA_F32_16X16X4_F32; ds_swizzle butterfly reduction;
// double-buffered partials -> single barrier per 16-step chunk.

#define Bsz 1024
#define Ssz 4096
#define Hsz 1024
#define NTHREADS 1024
#define NWAVES 32
#define CHUNK 16
#define NCHUNK (Ssz / CHUNK)          // 256
#define TROW 34                        // [t][h] tile row stride: 32 lanes + 2 pad (even -> b64-aligned)
#define TILE_PER_WAVE (CHUNK * TROW)   // 544 floats per wave
#define PART_STRIDE 33                 // 32 partials + 1 pad
#define PBUF_FLOATS (NWAVES * PART_STRIDE)

typedef float v2f __attribute__((ext_vector_type(2)));
typedef float v8f __attribute__((ext_vector_type(8)));

#define LOG2E 1.44269504088896340736f

#if __has_builtin(__builtin_amdgcn_exp2f)
#define EXP2F(x) __builtin_amdgcn_exp2f(x)
#else
#define EXP2F(x) exp2f(x)
#endif
#if __has_builtin(__builtin_amdgcn_rcpf)
#define RCPF(x) __builtin_amdgcn_rcpf(x)
#else
#define RCPF(x) (1.0f / (x))
#endif

// Deterministic wave32 butterfly sum via ds_swizzle group-of-32:
// offset = {0, xor[14:10], or[9:5], and[4:0]} -> SWAPX16/8/4/2/1.
__device__ __forceinline__ float wave_sum32(float v) {
    v += __int_as_float(__builtin_amdgcn_ds_swizzle(__float_as_int(v), 0x401f)); // xor 16
    v += __int_as_float(__builtin_amdgcn_ds_swizzle(__float_as_int(v), 0x201f)); // xor 8
    v += __int_as_float(__builtin_amdgcn_ds_swizzle(__float_as_int(v), 0x101f)); // xor 4
    v += __int_as_float(__builtin_amdgcn_ds_swizzle(__float_as_int(v), 0x081f)); // xor 2
    v += __int_as_float(__builtin_amdgcn_ds_swizzle(__float_as_int(v), 0x041f)); // xor 1
    return v;
}

__launch_bounds__(NTHREADS, 1)
__global__ void slstm_head_kernel(const float* __restrict__ x_codes,
                                  const float* __restrict__ Wi_w, const float* __restrict__ Wi_b,
                                  const float* __restrict__ Wf_w, const float* __restrict__ Wf_b,
                                  const float* __restrict__ Wo_w, const float* __restrict__ Wo_b,
                                  const float* __restrict__ Wg_w, const float* __restrict__ Wg_b,
                                  const float* __restrict__ proj_w, const float* __restrict__ proj_b,
                                  float* __restrict__ out)
{
    __shared__ float xs[Ssz];                               // 16 KB: pre-scaled x row
    __shared__ float htile[NWAVES * TILE_PER_WAVE];         // 68 KB: per-wave [t][h] tiles
    __shared__ float partial[2 * PBUF_FLOATS];              // 8.4 KB: double-buffered partials

    const int tid  = threadIdx.x;
    const int b    = blockIdx.x;
    const int lane = tid & 31;
    const int wave = tid >> 5;
    const int h    = tid;                                   // hidden index owned by thread

    // Stage + pre-scale x[b, :] into LDS (coalesced, 4 loads/thread).
    for (int i = tid; i < Ssz; i += NTHREADS)
        xs[i] = (x_codes[(size_t)b * Ssz + i] - 65.0f) * 0.01f;

    // Gate weights prescaled by log2(e): v_exp_f32 computes 2^x.
    // g-gate gets its sigmoid negation folded in as well.
    const float wiwl =  Wi_w[h] * LOG2E, wibl =  Wi_b[h] * LOG2E;
    const float wfwl =  Wf_w[h] * LOG2E, wfbl =  Wf_b[h] * LOG2E;
    const float wowl =  Wo_w[h] * LOG2E, wobl =  Wo_b[h] * LOG2E;
    const float wgwl = -Wg_w[h] * LOG2E, wgbl = -Wg_b[h] * LOG2E;
    const float NCOEF = 0.01f * LOG2E;                      // n kept in log2 domain
    const float pb = proj_b[wave & 1];                      // bias for this wave's (t,k) pair

    // Precompute WMMA B fragments: B is 4x16 (K=h slice, N=proj output col).
    // Lane L: n = L&15 ; V0 -> K = 2*(L>>4), V1 -> K = 2*(L>>4)+1. Only n<2 is live.
    const int nn = lane & 15;
    const int kb = (lane >> 4) * 2;
    v2f bfrag[8];
#pragma unroll
    for (int kk = 0; kk < 8; ++kk) {
        const int hg = wave * 32 + kk * 4 + kb;             // global h for K-slot
        bfrag[kk].x = (nn < 2) ? proj_w[nn * Hsz + hg]     : 0.0f;
        bfrag[kk].y = (nn < 2) ? proj_w[nn * Hsz + hg + 1] : 0.0f;
    }

    __syncthreads();

    float c = 0.0f, nl = 0.0f;                              // nl = n * log2(e)
    float* mytile = &htile[wave * TILE_PER_WAVE + lane];    // column = local h (this lane)
    const float* wtile = &htile[wave * TILE_PER_WAVE];

    for (int ch = 0; ch < NCHUNK; ++ch) {
        // Broadcast-read this chunk's 16 x values with 4x ds_load_b128.
        float xloc[CHUNK];
        {
            const float4* xp4 = (const float4*)(&xs[ch * CHUNK]);
#pragma unroll
            for (int q = 0; q < 4; ++q) {
                float4 v = xp4[q];
                xloc[4*q+0] = v.x; xloc[4*q+1] = v.y;
                xloc[4*q+2] = v.z; xloc[4*q+3] = v.w;
            }
        }

        // ---- 16 recurrence steps (serial chain; transcendental-bound) ----
#pragma unroll
        for (int tt = 0; tt < CHUNK; ++tt) {
            const float xv = xloc[tt];
            const float ai = fmaf(xv, wiwl, wibl);          // log2-domain pre-acts
            const float af = fmaf(xv, wfwl, wfbl);
            const float ao = fmaf(xv, wowl, wobl);
            const float ag = fmaf(xv, wgwl, wgbl);          // = -pre_g * log2(e)
            const float it = EXP2F(ai - nl);
            const float ft = EXP2F(af - nl);
            const float ot = EXP2F(ao - nl);
            const float gt = RCPF(1.0f + EXP2F(ag));
            c  = fmaf(ft, c, it * gt);
            nl = fmaf(NCOEF, (it + ft + ot - 3.0f), nl);
            const float sc = RCPF(1.0f + EXP2F(c * (-LOG2E)));
            mytile[tt * TROW] = sc * ot;                    // [t][h] stage, conflict-free
        }

        // Wave-private tile: ensure all lanes' DS stores landed before reads.
        asm volatile("s_wait_dscnt 0" ::: "memory");

        // ---- projection: D(16t x 16) += A(16t x 4h) * B(4h x 16), K = 32 per wave ----
        // A lane L: M(t) = L&15 ; V0 -> K = kb, V1 -> K = kb+1.
        // [t][h] layout + even stride => (a.x, a.y) is one aligned ds_load_b64.
        const int m = lane & 15;
        v8f acc = {};
#pragma unroll
        for (int kk = 0; kk < 8; ++kk) {
            const int row = kk * 4 + kb;                    // local h = K index (even)
            const v2f a = *(const v2f*)&wtile[m * TROW + row];
            acc = __builtin_amdgcn_wmma_f32_16x16x4_f32(
                false, a, false, bfrag[kk], (short)0, acc, false, false);
        }

        // C/D layout: VGPR r, lane L -> (t = r + 8*(L>=16), k = L&15). Live lanes: 0,1,16,17.
        float* pbuf = &partial[(ch & 1) * PBUF_FLOATS];     // double buffer, parity on chunk
        if ((lane & 15) < 2) {
            const int k    = lane & 15;
            const int toff = (lane >> 4) * 8;
#pragma unroll
            for (int r = 0; r < 8; ++r)
                pbuf[wave * PART_STRIDE + (toff + r) * 2 + k] = acc[r];
        }
        __syncthreads();                                    // single barrier per chunk

        // Distributed deterministic reduction: wave w owns (t,k) pair p = w.
        // Lane L reads wave L's partial for pair w (stride 33 -> conflict-free),
        // then a fixed-association ds_swizzle butterfly sums the 32 values.
        {
            const float v = wave_sum32(pbuf[lane * PART_STRIDE + wave]);
            if (lane == 0)
                out[(size_t)b * (Ssz * 2) + ch * 32 + wave] = v + pb;
        }
        // No second barrier needed: buffer (ch&1) is next rewritten in chunk ch+2,
        // which every wave reaches only after the chunk ch+1 barrier, and that
        // barrier orders it after all chunk-ch reads above.
    }
}

extern "C" void kernel_launch(void* const* d_in, const int* in_sizes, int n_in,
                              void* d_out, int out_size, void* d_ws, size_t ws_size,
                              hipStream_t stream) {
    (void)in_sizes; (void)n_in; (void)d_ws; (void)ws_size; (void)out_size;
    const float* x_codes = (const float*)d_in[0];
    const float* Wi_w = (const float*)d_in[1];
    const float* Wi_b = (const float*)d_in[2];
    const float* Wf_w = (const float*)d_in[3];
    const float* Wf_b = (const float*)d_in[4];
    const float* Wo_w = (const float*)d_in[5];
    const float* Wo_b = (const float*)d_in[6];
    const float* Wg_w = (const float*)d_in[7];
    const float* Wg_b = (const float*)d_in[8];
    const float* proj_w = (const float*)d_in[9];
    const float* proj_b = (const float*)d_in[10];
    float* out = (float*)d_out;

    slstm_head_kernel<<<dim3(Bsz), dim3(NTHREADS), 0, stream>>>(
        x_codes, Wi_w, Wi_b, Wf_w, Wf_b, Wo_w, Wo_b, Wg_w, Wg_b, proj_w, proj_b, out);
}